// GCN2_47124381171999
// MI455X (gfx1250) — compile-verified
//
#include <hip/hip_runtime.h>
#include <hip/hip_bf16.h>

typedef float v2f __attribute__((ext_vector_type(2)));
typedef float v8f __attribute__((ext_vector_type(8)));

#define N_NODES   50000
#define N_EDGES   800000
#define FEATS     128
#define N_GRAPHS  64
#define N_CLASSES 10

// ---------------- utility: zero a float region (float4 granularity) ----------
__global__ void GCN2_zero_kernel(float* __restrict__ p, int n4) {
    int i = blockIdx.x * blockDim.x + threadIdx.x;
    if (i < n4) ((float4*)p)[i] = make_float4(0.f, 0.f, 0.f, 0.f);
}

// ---------------- degrees via atomics --------------------------------------
__global__ void GCN2_degree_kernel(const int* __restrict__ src, const int* __restrict__ dst,
                                   float* __restrict__ degO, float* __restrict__ degI) {
    int e = blockIdx.x * blockDim.x + threadIdx.x;
    if (e < N_EDGES) {
        atomicAdd(degO + src[e], 1.0f);
        atomicAdd(degI + dst[e], 1.0f);
    }
}

// deg -> rsqrt(max(deg,1)) in place, over both arrays (contiguous)
__global__ void GCN2_norm_kernel(float* __restrict__ deg, int n) {
    int i = blockIdx.x * blockDim.x + threadIdx.x;
    if (i < n) deg[i] = rsqrtf(fmaxf(deg[i], 1.0f));
}

// ---------------- WMMA GEMM: out[row,:] = (A[row,:] @ W) * rowscale[row] -----
// A: [N_NODES,128] row-major, W: [128,128] row-major.
// Block = 128 threads = 4 waves; each wave computes a 16x128 strip using
// V_WMMA_F32_16X16X4_F32. W staged in LDS with K-pairs swizzled adjacent so a
// B fragment is one ds_load_b64.
__global__ __launch_bounds__(128) void GCN2_gemm_rowscale_kernel(
        const float* __restrict__ A, const float* __restrict__ W,
        const float* __restrict__ rowscale, float* __restrict__ out) {
    __shared__ float lw[FEATS * FEATS];   // 64 KB, layout: [(k>>1)*256 + col*2 + (k&1)]
    int tid = threadIdx.x;
    for (int i = tid; i < FEATS * FEATS; i += 128) {
        int k = i >> 7, c = i & 127;
        lw[((k >> 1) << 8) + (c << 1) + (k & 1)] = W[i];
    }
    __syncthreads();

    int wave = tid >> 5, lane = tid & 31;
    int row0 = (blockIdx.x * 4 + wave) << 4;      // 16-row tile per wave
    if (row0 >= N_NODES) return;                  // N_NODES % 16 == 0: whole-wave guard

    int hi = lane >> 4;        // 0: K pair {0,1}; 1: K pair {2,3}
    int lo = lane & 15;

    v8f acc[8] = {};           // 8 N-tiles of 16 cols -> full 128 cols

    // A frag layout (f32 16x4): lanes 0-15 hold K=k0+0/k0+1, lanes 16-31 K=k0+2/k0+3
    const float* aRow = A + (size_t)(row0 + lo) * FEATS + (hi << 1);

    for (int k0 = 0; k0 < FEATS; k0 += 4) {
        v2f afrag = *(const v2f*)(aRow + k0);
        const float* bk = lw + ((((k0 >> 1) + hi)) << 8) + (lo << 1);
#pragma unroll
        for (int t = 0; t < 8; t++) {
            // B frag (4x16): VGPR0 = W[k][n0+lo], VGPR1 = W[k+1][n0+lo]; adjacent in LDS
            v2f bfrag = *(const v2f*)(bk + (t << 5));
            acc[t] = __builtin_amdgcn_wmma_f32_16x16x4_f32(
                false, afrag, false, bfrag, (short)0, acc[t], false, false);
        }
    }

    // C/D layout: VGPR r, lanes 0-15 -> row=row0+r, lanes 16-31 -> row=row0+8+r; col = n0+lo
#pragma unroll
    for (int r = 0; r < 8; r++) {
        int row = row0 + r + (hi << 3);
        float s = rowscale[row];
        float* o = out + (size_t)row * FEATS + lo;
#pragma unroll
        for (int t = 0; t < 8; t++) o[t << 4] = acc[t][r] * s;
    }
}

// ---------------- edge scatter: agg[dst] += h[src], one wave per edge -------
__global__ __launch_bounds__(256) void GCN2_scatter_kernel(
        const int* __restrict__ src, const int* __restrict__ dst,
        const float* __restrict__ h, float* __restrict__ agg) {
    int gid = blockIdx.x * blockDim.x + threadIdx.x;
    int e = gid >> 5;
    if (e >= N_EDGES) return;
    int lane = threadIdx.x & 31;
    int s = src[e], d = dst[e];
    const float4 v = *(const float4*)(h + (size_t)s * FEATS + (lane << 2));
    float* o = agg + (size_t)d * FEATS + (lane << 2);
    atomicAdd(o + 0, v.x);
    atomicAdd(o + 1, v.y);
    atomicAdd(o + 2, v.z);
    atomicAdd(o + 3, v.w);
}

// ---------------- finalize: out = [relu](agg * norm_dst[row] + b[col]) ------
__global__ void GCN2_finalize_kernel(const float* __restrict__ agg,
                                     const float* __restrict__ normd,
                                     const float* __restrict__ bias,
                                     float* __restrict__ out, int relu) {
    int i = blockIdx.x * blockDim.x + threadIdx.x;   // float4 index
    if (i >= N_NODES * (FEATS / 4)) return;
    int row = i >> 5;          // 32 float4 per row
    int c4  = i & 31;
    float nd = normd[row];
    float4 v = ((const float4*)agg)[i];
    float4 b = ((const float4*)bias)[c4];
    v.x = v.x * nd + b.x; v.y = v.y * nd + b.y;
    v.z = v.z * nd + b.z; v.w = v.w * nd + b.w;
    if (relu) {
        v.x = fmaxf(v.x, 0.f); v.y = fmaxf(v.y, 0.f);
        v.z = fmaxf(v.z, 0.f); v.w = fmaxf(v.w, 0.f);
    }
    ((float4*)out)[i] = v;
}

// ---------------- pooling ----------------------------------------------------
__global__ void GCN2_pool_sum_kernel(const float* __restrict__ h,
                                     const int* __restrict__ gids,
                                     float* __restrict__ hg) {
    int i = blockIdx.x * blockDim.x + threadIdx.x;
    if (i >= N_NODES * FEATS) return;
    int node = i >> 7, c = i & 127;
    atomicAdd(hg + gids[node] * FEATS + c, h[i]);
}

__global__ void GCN2_count_kernel(const int* __restrict__ gids, float* __restrict__ cnt) {
    int i = blockIdx.x * blockDim.x + threadIdx.x;
    if (i < N_NODES) atomicAdd(cnt + gids[i], 1.0f);
}

// ---------------- classifier: out[g,cls] = (hg_sum[g,:]/cnt) @ Wc + bc -------
__global__ __launch_bounds__(640) void GCN2_classify_kernel(
        const float* __restrict__ hg, const float* __restrict__ cnt,
        const float* __restrict__ Wc, const float* __restrict__ bc,
        float* __restrict__ out) {
    int t = threadIdx.x;
    if (t >= N_GRAPHS * N_CLASSES) return;
    int g = t / N_CLASSES, cls = t % N_CLASSES;
    float inv = 1.0f / fmaxf(cnt[g], 1.0f);
    float acc = 0.f;
    for (int k = 0; k < FEATS; k++)
        acc += hg[g * FEATS + k] * Wc[k * N_CLASSES + cls];
    out[t] = acc * inv + bc[cls];
}

extern "C" void kernel_launch(void* const* d_in, const int* in_sizes, int n_in,
                              void* d_out, int out_size, void* d_ws, size_t ws_size,
                              hipStream_t stream) {
    (void)in_sizes; (void)n_in; (void)out_size; (void)ws_size;
    const int*   src  = (const int*)d_in[0];
    const int*   dst  = (const int*)d_in[1];
    const int*   gids = (const int*)d_in[2];
    const float* x    = (const float*)d_in[3];
    const float* W1   = (const float*)d_in[4];
    const float* b1   = (const float*)d_in[5];
    const float* W2   = (const float*)d_in[6];
    const float* b2   = (const float*)d_in[7];
    const float* Wc   = (const float*)d_in[8];
    const float* bc   = (const float*)d_in[9];
    float* out = (float*)d_out;

    float* ws   = (float*)d_ws;
    float* ns   = ws;                                  // 50000  (deg_out -> norm_src)
    float* nd   = ns + N_NODES;                        // 50000  (deg_in  -> norm_dst)
    float* bufA = nd + N_NODES;                        // 50000*128
    float* bufB = bufA + (size_t)N_NODES * FEATS;      // 50000*128
    float* hg   = bufB + (size_t)N_NODES * FEATS;      // 64*128
    // cnt (64 floats) immediately follows hg

    const int NF4   = N_NODES * FEATS / 4;             // 1.6M float4
    const int gemmB = (N_NODES / 16 + 3) / 4;          // 782 blocks (4 waves each)

    // degrees + norms
    GCN2_zero_kernel<<<(2 * N_NODES / 4 + 255) / 256, 256, 0, stream>>>(ns, 2 * N_NODES / 4);
    GCN2_degree_kernel<<<(N_EDGES + 255) / 256, 256, 0, stream>>>(src, dst, ns, nd);
    GCN2_norm_kernel<<<(2 * N_NODES + 255) / 256, 256, 0, stream>>>(ns, 2 * N_NODES);

    // layer 1: bufA = (x @ W1) * ns ; bufB = scatter ; bufA = relu(bufB*nd + b1)
    GCN2_gemm_rowscale_kernel<<<gemmB, 128, 0, stream>>>(x, W1, ns, bufA);
    GCN2_zero_kernel<<<(NF4 + 255) / 256, 256, 0, stream>>>(bufB, NF4);
    GCN2_scatter_kernel<<<(N_EDGES * 32 + 255) / 256, 256, 0, stream>>>(src, dst, bufA, bufB);
    GCN2_finalize_kernel<<<(NF4 + 255) / 256, 256, 0, stream>>>(bufB, nd, b1, bufA, 1);

    // layer 2: bufB = (bufA @ W2) * ns ; bufA = scatter ; bufB = bufA*nd + b2
    GCN2_gemm_rowscale_kernel<<<gemmB, 128, 0, stream>>>(bufA, W2, ns, bufB);
    GCN2_zero_kernel<<<(NF4 + 255) / 256, 256, 0, stream>>>(bufA, NF4);
    GCN2_scatter_kernel<<<(N_EDGES * 32 + 255) / 256, 256, 0, stream>>>(src, dst, bufB, bufA);
    GCN2_finalize_kernel<<<(NF4 + 255) / 256, 256, 0, stream>>>(bufA, nd, b2, bufB, 0);

    // pooling + classifier
    int poolZ = (N_GRAPHS * FEATS + N_GRAPHS) / 4;     // hg + cnt contiguous
    GCN2_zero_kernel<<<(poolZ + 255) / 256, 256, 0, stream>>>(hg, poolZ);
    GCN2_pool_sum_kernel<<<(N_NODES * FEATS + 255) / 256, 256, 0, stream>>>(bufB, gids, hg);
    GCN2_count_kernel<<<(N_NODES + 255) / 256, 256, 0, stream>>>(gids, hg + N_GRAPHS * FEATS);
    GCN2_classify_kernel<<<1, 640, 0, stream>>>(hg, hg + N_GRAPHS * FEATS, Wc, bc, out);
}